// ProbAttention_57604101374008
// MI455X (gfx1250) — compile-verified
//
#include <hip/hip_runtime.h>
#include <math.h>

typedef __attribute__((ext_vector_type(2))) float v2f;
typedef __attribute__((ext_vector_type(8))) float v8f;

#define B_ 8
#define L_ 4096
#define H_ 8
#define D_ 64
#define CHUNK 256
#define NCHUNK (L_ / CHUNK)

// ---------------------------------------------------------------------------
// Kernel 1: sparsity measure M[b,h,q] = max_s(QK_s) - sum_s(QK_s)/L
// one wave32 per query; D=64 -> 2 elements per lane
// ---------------------------------------------------------------------------
__global__ __launch_bounds__(256) void probM_kernel(
    const float* __restrict__ Q, const float* __restrict__ K,
    const int* __restrict__ idxs, float* __restrict__ M, int S) {
  int wave = threadIdx.x >> 5;
  int lane = threadIdx.x & 31;
  int qg = blockIdx.x * 8 + wave;           // global query id over B*H*L
  int bh = qg >> 12;                        // / L_ (L_ = 4096)
  int q  = qg & (L_ - 1);
  int b = bh / H_, h = bh % H_;
  const float* qrow = Q + (((long long)(b * L_ + q)) * H_ + h) * D_;
  float q0 = qrow[lane], q1 = qrow[lane + 32];
  float mx = -3.0e38f, sm = 0.0f;
  for (int s = 0; s < S; ++s) {
    int ki = idxs[q * S + s];
    const float* krow = K + (((long long)(b * L_ + ki)) * H_ + h) * D_;
    float p = q0 * krow[lane] + q1 * krow[lane + 32];
#pragma unroll
    for (int off = 16; off >= 1; off >>= 1) p += __shfl_xor(p, off, 32);
    mx = fmaxf(mx, p);
    sm += p;
  }
  if (lane == 0) M[(long long)bh * L_ + q] = mx - sm / (float)L_;
}

// ---------------------------------------------------------------------------
// Kernel 2: top-U indices of M per (b,h) via iterative masked argmax in LDS
// ---------------------------------------------------------------------------
__global__ __launch_bounds__(256) void topk_kernel(
    const float* __restrict__ M, int* __restrict__ top, int U) {
  __shared__ float vals[L_];
  __shared__ float rv[256];
  __shared__ int   ri[256];
  int bh = blockIdx.x, tid = threadIdx.x;
  const float* m = M + (long long)bh * L_;
  for (int i = tid; i < L_; i += 256) vals[i] = m[i];
  __syncthreads();
  for (int it = 0; it < U; ++it) {
    float bv = -3.0e38f; int bi = 0;
    for (int i = tid; i < L_; i += 256) {
      float v = vals[i];
      if (v > bv) { bv = v; bi = i; }        // strict > keeps lowest index on tie
    }
    rv[tid] = bv; ri[tid] = bi;
    __syncthreads();
    for (int st = 128; st > 0; st >>= 1) {
      if (tid < st) {
        float ov = rv[tid + st]; int oi = ri[tid + st];
        if (ov > rv[tid] || (ov == rv[tid] && oi < ri[tid])) { rv[tid] = ov; ri[tid] = oi; }
      }
      __syncthreads();
    }
    if (tid == 0) { top[bh * U + it] = ri[0]; vals[ri[0]] = -3.0e38f; }
    __syncthreads();
  }
}

// ---------------------------------------------------------------------------
// Kernels 3+4: inclusive cumsum of V along L, written to out (B,L,H,D layout)
// chunked scan, 4x unrolled so each wave keeps several loads in flight
// ---------------------------------------------------------------------------
__global__ __launch_bounds__(64) void vsum_kernel(
    const float* __restrict__ V, float* __restrict__ part) {
  int blk = blockIdx.x;
  int bh = blk / NCHUNK, c = blk % NCHUNK;
  int b = bh / H_, h = bh % H_;
  int d = threadIdx.x;
  int l0 = c * CHUNK;
  const float* base = V + (((long long)(b * L_ + l0)) * H_ + h) * D_ + d;
  const long long st = (long long)H_ * D_;
  float acc = 0.0f;
  for (int i = 0; i < CHUNK; i += 4) {
    float a0 = base[(i + 0) * st];
    float a1 = base[(i + 1) * st];
    float a2 = base[(i + 2) * st];
    float a3 = base[(i + 3) * st];
    acc += (a0 + a1) + (a2 + a3);
  }
  part[((long long)bh * NCHUNK + c) * D_ + d] = acc;
}

__global__ __launch_bounds__(64) void cumsum_kernel(
    const float* __restrict__ V, const float* __restrict__ part,
    float* __restrict__ out) {
  int blk = blockIdx.x;
  int bh = blk / NCHUNK, c = blk % NCHUNK;
  int b = bh / H_, h = bh % H_;
  int d = threadIdx.x;
  float acc = 0.0f;
  for (int cc = 0; cc < c; ++cc) acc += part[((long long)bh * NCHUNK + cc) * D_ + d];
  int l0 = c * CHUNK;
  const float* vb = V + (((long long)(b * L_ + l0)) * H_ + h) * D_ + d;
  float*       ob = out + (((long long)(b * L_ + l0)) * H_ + h) * D_ + d;
  const long long st = (long long)H_ * D_;
  for (int i = 0; i < CHUNK; i += 4) {
    float a0 = vb[(i + 0) * st];
    float a1 = vb[(i + 1) * st];
    float a2 = vb[(i + 2) * st];
    float a3 = vb[(i + 3) * st];
    acc += a0; ob[(i + 0) * st] = acc;
    acc += a1; ob[(i + 1) * st] = acc;
    acc += a2; ob[(i + 2) * st] = acc;
    acc += a3; ob[(i + 3) * st] = acc;
  }
}

// ---------------------------------------------------------------------------
// Kernel 5: split-K flash attention partials for the U selected queries.
// grid = (b,h,split); block = 96 threads = 3 waves; wave w owns q-rows
// [16w,16w+16). Each split covers L/nsplit keys, online softmax over 16-key
// tiles with fp32 WMMA 16x16x4 for QK^T and PV. Emits unnormalized (m,l,O).
// ---------------------------------------------------------------------------
__global__ __launch_bounds__(96) void probattn_part_kernel(
    const float* __restrict__ Q, const float* __restrict__ K,
    const float* __restrict__ V, const int* __restrict__ top,
    float* __restrict__ wsML, float* __restrict__ wsO, int U, int nsplit) {
  __shared__ float sQ[48][68];     // padded to dodge bank conflicts
  __shared__ float sK[16][68];
  __shared__ float sV[16][68];
  __shared__ float sP[3][16][20];
  __shared__ int   sQpos[48];

  int blk = blockIdx.x;
  int sp = blk % nsplit, bh = blk / nsplit;
  int b = bh / H_, h = bh % H_;
  int splitlen = L_ / nsplit;
  int key_lo = sp * splitlen;
  int tid = threadIdx.x, lane = tid & 31, wv = tid >> 5;
  int ln = lane & 15;
  int off2 = (lane & 16) ? 2 : 0;
  int halfrow = (lane & 16) ? 8 : 0;
  const int qt = wv;

  if (tid < 48) {
    int i = tid < U ? tid : (U - 1);        // pad rows reuse a real index (discarded)
    sQpos[tid] = top[bh * U + i];
  }
  __syncthreads();
  int maxq = 0;
  for (int i = 0; i < 48; ++i) maxq = max(maxq, sQpos[i]);

  // load Q_reduce rows (48 x 64) via 16B segments
  for (int i = tid; i < 48 * 16; i += 96) {
    int r = i >> 4, sg = i & 15;
    const float4 v = *(const float4*)(Q + (((long long)(b * L_ + sQpos[r])) * H_ + h) * D_ + sg * 4);
    sQ[r][sg * 4 + 0] = v.x; sQ[r][sg * 4 + 1] = v.y;
    sQ[r][sg * 4 + 2] = v.z; sQ[r][sg * 4 + 3] = v.w;
  }

  float m_i[8], l_i[8];
  v8f acc[4];
#pragma unroll
  for (int r = 0; r < 8; ++r) { m_i[r] = -3.0e38f; l_i[r] = 0.0f; }
#pragma unroll
  for (int t = 0; t < 4; ++t) { v8f z = {}; acc[t] = z; }

  int ntile = splitlen / 16;
  for (int kb = 0; kb < ntile; ++kb) {
    int key0 = key_lo + kb * 16;
    if (key0 > maxq) break;                 // fully masked tail (uniform branch)
    __syncthreads();
    // cooperative K/V tile load (16 keys x 64)
    for (int i = tid; i < 256; i += 96) {
      int r = i >> 4, sg = i & 15;
      long long rowoff = (((long long)(b * L_ + key0 + r)) * H_ + h) * D_ + sg * 4;
      float4 kv = *(const float4*)(K + rowoff);
      float4 vv = *(const float4*)(V + rowoff);
      sK[r][sg * 4 + 0] = kv.x; sK[r][sg * 4 + 1] = kv.y;
      sK[r][sg * 4 + 2] = kv.z; sK[r][sg * 4 + 3] = kv.w;
      sV[r][sg * 4 + 0] = vv.x; sV[r][sg * 4 + 1] = vv.y;
      sV[r][sg * 4 + 2] = vv.z; sV[r][sg * 4 + 3] = vv.w;
    }
    if (kb + 1 < ntile) {        // hint the next K tile (global_prefetch_b8)
      int r = tid & 15;
      const float* nk = K + (((long long)(b * L_ + key0 + 16 + r)) * H_ + h) * D_ + (tid >> 4) * 10;
      __builtin_prefetch(nk, 0, 0);
    }
    __syncthreads();

    // S = Qr * K^T  (16 WMMAs over d)
    v8f s = {};
#pragma unroll
    for (int dd = 0; dd < 16; ++dd) {
      v2f a, bb;
      a.x  = sQ[qt * 16 + ln][4 * dd + off2];
      a.y  = sQ[qt * 16 + ln][4 * dd + off2 + 1];
      bb.x = sK[ln][4 * dd + off2];
      bb.y = sK[ln][4 * dd + off2 + 1];
      s = __builtin_amdgcn_wmma_f32_16x16x4_f32(false, a, false, bb, (short)0, s, false, false);
    }

    // scale, causal mask, online softmax (row = component r + halfrow)
#pragma unroll
    for (int r = 0; r < 8; ++r) {
      int row  = r + halfrow;
      int qpos = sQpos[qt * 16 + row];
      int key  = key0 + ln;
      float sv = s[r] * 0.125f;               // 1/sqrt(64)
      if (key > qpos) sv = -3.0e38f;
      float rmax = sv;
#pragma unroll
      for (int o = 8; o >= 1; o >>= 1) rmax = fmaxf(rmax, __shfl_xor(rmax, o, 16));
      float nm = fmaxf(m_i[r], rmax);
      float p = __expf(sv - nm);
      float rsum = p;
#pragma unroll
      for (int o = 8; o >= 1; o >>= 1) rsum += __shfl_xor(rsum, o, 16);
      float corr = __expf(m_i[r] - nm);
      l_i[r] = l_i[r] * corr + rsum;
      m_i[r] = nm;
#pragma unroll
      for (int t = 0; t < 4; ++t) acc[t][r] *= corr;
      sP[qt][row][ln] = p;                    // C-layout -> LDS for A-layout reload
    }
    __syncthreads();

    // O += P * V  (4 d-tiles x 4 k-chunks of WMMA)
#pragma unroll
    for (int t = 0; t < 4; ++t) {
      v8f c = acc[t];
#pragma unroll
      for (int kk = 0; kk < 4; ++kk) {
        v2f a, bb;
        a.x  = sP[qt][ln][4 * kk + off2];
        a.y  = sP[qt][ln][4 * kk + off2 + 1];
        bb.x = sV[4 * kk + off2][t * 16 + ln];
        bb.y = sV[4 * kk + off2 + 1][t * 16 + ln];
        c = __builtin_amdgcn_wmma_f32_16x16x4_f32(false, a, false, bb, (short)0, c, false, false);
      }
      acc[t] = c;
    }
  }

  // emit unnormalized partials: (m, l) per row, O per row x d
#pragma unroll
  for (int r = 0; r < 8; ++r) {
    int row = r + halfrow;
    long long idx = ((long long)blk * 48 + qt * 16 + row);
    if (ln == 0) {
      wsML[idx * 2 + 0] = m_i[r];
      wsML[idx * 2 + 1] = l_i[r];
    }
#pragma unroll
    for (int t = 0; t < 4; ++t) wsO[idx * D_ + t * 16 + ln] = acc[t][r];
  }
}

// ---------------------------------------------------------------------------
// Kernel 6: merge split-K partials, normalize, scatter over cumsum baseline
// grid = (b,h,q<U); block = 64 threads (one per d)
// ---------------------------------------------------------------------------
__global__ __launch_bounds__(64) void probattn_merge_kernel(
    const float* __restrict__ wsML, const float* __restrict__ wsO,
    const int* __restrict__ top, float* __restrict__ out, int U, int nsplit) {
  int blk = blockIdx.x;
  int bh = blk / U, q = blk % U;
  int b = bh / H_, h = bh % H_;
  int d = threadIdx.x;
  float M = -3.0e38f;
  for (int s = 0; s < nsplit; ++s) {
    long long idx = (((long long)bh * nsplit + s) * 48 + q);
    M = fmaxf(M, wsML[idx * 2 + 0]);
  }
  float Lsum = 0.0f, o = 0.0f;
  for (int s = 0; s < nsplit; ++s) {
    long long idx = (((long long)bh * nsplit + s) * 48 + q);
    float w = __expf(wsML[idx * 2 + 0] - M);
    Lsum += wsML[idx * 2 + 1] * w;
    o    += wsO[idx * D_ + d] * w;
  }
  int qpos = top[bh * U + q];
  out[(((long long)(b * L_ + qpos)) * H_ + h) * D_ + d] = o / Lsum;
}

// ---------------------------------------------------------------------------
extern "C" void kernel_launch(void* const* d_in, const int* in_sizes, int n_in,
                              void* d_out, int out_size, void* d_ws, size_t ws_size,
                              hipStream_t stream) {
  const float* Q = (const float*)d_in[0];
  const float* K = (const float*)d_in[1];
  const float* V = (const float*)d_in[2];
  const int* idxs = (const int*)d_in[3];
  float* out = (float*)d_out;

  int S = in_sizes[3] / L_;     // sample_k (= 45)
  int U = S;                    // u = FACTOR*ceil(log L) == sample_k here
  if (U > 48) U = 48;

  char* ws = (char*)d_ws;
  size_t o = 0;
  float* wsM = (float*)(ws + o);                             // B*H*L floats
  o += (size_t)B_ * H_ * L_ * sizeof(float);
  int* wsTop = (int*)(ws + o);
  o += (size_t)B_ * H_ * U * sizeof(int);
  o = (o + 255) & ~(size_t)255;
  float* wsPart = (float*)(ws + o);                          // B*H*NCHUNK*D floats
  o += (size_t)B_ * H_ * NCHUNK * D_ * sizeof(float);
  o = (o + 255) & ~(size_t)255;

  // pick the largest split count whose partial buffers fit in the workspace
  int ns = 8;
  for (;;) {
    size_t need = o
      + (size_t)B_ * H_ * ns * 48 * 2 * sizeof(float)   // (m,l)
      + (size_t)B_ * H_ * ns * 48 * D_ * sizeof(float)  // O
      + 256;
    if (need <= ws_size || ns == 1) break;
    ns >>= 1;
  }
  float* wsML = (float*)(ws + o);
  o += (size_t)B_ * H_ * ns * 48 * 2 * sizeof(float);
  o = (o + 255) & ~(size_t)255;
  float* wsO = (float*)(ws + o);

  probM_kernel<<<B_ * H_ * L_ / 8, 256, 0, stream>>>(Q, K, idxs, wsM, S);
  topk_kernel<<<B_ * H_, 256, 0, stream>>>(wsM, wsTop, U);
  vsum_kernel<<<B_ * H_ * NCHUNK, 64, 0, stream>>>(V, wsPart);
  cumsum_kernel<<<B_ * H_ * NCHUNK, 64, 0, stream>>>(V, wsPart, out);
  probattn_part_kernel<<<B_ * H_ * ns, 96, 0, stream>>>(Q, K, V, wsTop, wsML, wsO, U, ns);
  probattn_merge_kernel<<<B_ * H_ * U, 64, 0, stream>>>(wsML, wsO, wsTop, out, U, ns);
}